// LowpassFilter_38173669327225
// MI455X (gfx1250) — compile-verified
//
#include <hip/hip_runtime.h>

// ---------------- problem constants ----------------
#define BATCH   64
#define FREQ    512
#define FRAMES  64
#define WIN     1022
#define HOP     256
#define SIG_LEN 17150
#define PADL    18
#define EXT_LEN (SIG_LEN + 2 * PADL)   // 17186
#define MROWS   (BATCH * FRAMES)       // 4096
#define KDIM    1024                   // padded K (512 re + 512 im, or 1022 window)
#define NDIM    1024                   // padded N

#define TWO_PI_OVER_N ((float)(6.283185307179586476925286766559 / 1022.0))

// Butterworth(5, 0.5) digital coefficients (exact closed form):
// poles at 0, +-j*tan(18deg), +-j*tan(36deg)
#define FB0 0.0527864045f
#define FB1 0.2639320225f
#define FB2 0.5278640450f
#define FA2 0.6334368540f   // tan^2(18) + tan^2(36)
#define FA4 0.0557280900f   // tan^2(18) * tan^2(36)
// lfilter_zi (steady-state initial conditions)
#define ZI0 0.9472135955f
#define ZI1 0.6832815730f
#define ZI2 0.7888543820f
#define ZI3 0.2609903044f
#define ZI4 0.0527864045f

// IIR segment-parallel scan: poles |z|=0.7265 -> state decays 0.7265^96 ~ 3e-14,
// so a 96-sample zero-state warm-up reproduces the exact scan to f32 precision.
#define SEG   256
#define WARM  96
#define NSEG  ((EXT_LEN + SEG - 1) / SEG)   // 68

typedef float v2f __attribute__((ext_vector_type(2)));
typedef float v8f __attribute__((ext_vector_type(8)));

// ---------------- window setup (1 block) ----------------
__global__ __launch_bounds__(256) void build_windows(float* __restrict__ hann,
                                                     float* __restrict__ syn) {
  __shared__ float denom[HOP];
  const int j = threadIdx.x;
  float d = 0.f;
  #pragma unroll
  for (int o = 0; o < 4; ++o) {
    int idx = o * HOP + j;
    if (idx < WIN) {
      float w = 0.5f - 0.5f * cosf((float)idx * TWO_PI_OVER_N);
      d += w * w;
    }
  }
  denom[j] = d;
  __syncthreads();
  for (int n = j; n < NDIM; n += 256) {
    float w = (n < WIN) ? (0.5f - 0.5f * cosf((float)n * TWO_PI_OVER_N)) : 0.f;
    hann[n] = w;
    syn[n]  = (n < WIN) ? (w / denom[n & (HOP - 1)]) : 0.f;
  }
}

// ---------------- iSTFT matrix: W1[k][n] ----------------
// k<512 : wk*cos(2*pi*k*n/1022)/1022 * syn[n]
// k>=512: -wk*sin(2*pi*(k-512)*n/1022)/1022 * syn[n],  wk = 1 for k in {0,511} else 2
__global__ __launch_bounds__(256) void build_W1(float* __restrict__ W1,
                                                const float* __restrict__ syn) {
  int flat = blockIdx.x * 256 + threadIdx.x;      // 1024*1024 entries
  int k = flat >> 10, n = flat & 1023;
  float v = 0.f;
  if (n < WIN) {
    int kk = (k >= 512) ? (k - 512) : k;
    int m  = (kk * n) % WIN;                      // exact angle reduction
    float ang = (float)m * TWO_PI_OVER_N;
    float wk  = (kk == 0 || kk == 511) ? 1.f : 2.f;
    float base = wk * (1.0f / 1022.0f) * syn[n];
    v = (k < 512) ? base * cosf(ang) : -base * sinf(ang);
  }
  W1[flat] = v;
}

// ---------------- STFT matrix: W2[n][c] (Hann folded into rows) ----------------
__global__ __launch_bounds__(256) void build_W2(float* __restrict__ W2,
                                                const float* __restrict__ hann) {
  int flat = blockIdx.x * 256 + threadIdx.x;
  int n = flat >> 10, c = flat & 1023;
  float v = 0.f;
  if (n < WIN) {
    int cc = c & 511;
    int m  = (cc * n) % WIN;
    float ang = (float)m * TWO_PI_OVER_N;
    float h = hann[n];
    v = (c < 512) ? h * cosf(ang) : -h * sinf(ang);
  }
  W2[flat] = v;
}

// ---------------- pack input [64,512,64,2] -> X[4096][1024] (re | im) ----------------
__global__ __launch_bounds__(256) void pack_X(const float* __restrict__ in,
                                              float* __restrict__ X) {
  int flat = blockIdx.x * 256 + threadIdx.x;      // 4096*1024
  int r = flat >> 10, k = flat & 1023;
  int b = r >> 6, f = r & 63;
  int kk = k & 511, p = k >> 9;
  X[flat] = in[((b * FREQ + kk) * FRAMES + f) * 2 + p];
}

// ---------------- f32 WMMA GEMM: C[4096,1024] = A[4096,1024] * W[1024,1024] ----------------
// 128 threads = 4 waves (2x2); wave tile 32x64 = 2 A-frags x 4 B-frags, 8 accumulators.
// W staged TRANSPOSED in LDS so both A and B fragments are single ds_load_b64.
__global__ __launch_bounds__(128) void gemm_wmma_f32(const float* __restrict__ A,
                                                     const float* __restrict__ W,
                                                     float* __restrict__ C) {
  __shared__ float As[64][34];     // [m][k]  (stride 34: 8B-aligned rows, bank spread)
  __shared__ float Wt[128][34];    // [n][k]  transposed
  const int tid  = threadIdx.x;
  const int m0   = blockIdx.y * 64;
  const int n0   = blockIdx.x * 128;
  const int wave = tid >> 5;
  const int lane = tid & 31;
  const int l16  = lane & 15;
  const int lh   = lane >> 4;
  const int wM   = (wave >> 1) * 32;   // 0 or 32
  const int wN   = (wave & 1) * 64;    // 0 or 64

  v8f acc[2][4];
  #pragma unroll
  for (int a = 0; a < 2; ++a)
    #pragma unroll
    for (int t = 0; t < 4; ++t)
      acc[a][t] = (v8f){0.f,0.f,0.f,0.f,0.f,0.f,0.f,0.f};

  for (int kb = 0; kb < KDIM; kb += 32) {
    // stage A tile 64x32 (512 float4, 4 per thread)
    #pragma unroll
    for (int i = 0; i < 4; ++i) {
      int idx = tid + i * 128;
      int row = idx >> 3;
      int c4  = (idx & 7) << 2;
      float4 v = *reinterpret_cast<const float4*>(&A[(m0 + row) * KDIM + kb + c4]);
      As[row][c4 + 0] = v.x; As[row][c4 + 1] = v.y;
      As[row][c4 + 2] = v.z; As[row][c4 + 3] = v.w;
    }
    // stage W tile 32x128 transposed (1024 float4, 8 per thread)
    #pragma unroll
    for (int i = 0; i < 8; ++i) {
      int idx = tid + i * 128;
      int kr  = idx >> 5;
      int n4  = (idx & 31) << 2;
      float4 v = *reinterpret_cast<const float4*>(&W[(kb + kr) * NDIM + n0 + n4]);
      Wt[n4 + 0][kr] = v.x; Wt[n4 + 1][kr] = v.y;
      Wt[n4 + 2][kr] = v.z; Wt[n4 + 3][kr] = v.w;
    }
    __syncthreads();
    if (kb + 32 < KDIM) {   // global_prefetch_b8 next K tile
      __builtin_prefetch(&A[(m0 + (tid >> 3) + 32) * KDIM + kb + 32], 0, 1);
      __builtin_prefetch(&A[(m0 + (tid >> 3)) * KDIM + kb + 32], 0, 1);
      __builtin_prefetch(&W[(kb + 32 + (tid >> 5)) * NDIM + n0 + ((tid & 31) << 2)], 0, 1);
    }
    #pragma unroll
    for (int kk = 0; kk < 32; kk += 4) {
      const int ks = kk + 2 * lh;   // lanes 0-15: k,k+1 ; lanes 16-31: k+2,k+3
      v2f a0 = *reinterpret_cast<const v2f*>(&As[wM + l16][ks]);
      v2f a1 = *reinterpret_cast<const v2f*>(&As[wM + 16 + l16][ks]);
      #pragma unroll
      for (int t = 0; t < 4; ++t) {
        v2f b = *reinterpret_cast<const v2f*>(&Wt[wN + t * 16 + l16][ks]);
        acc[0][t] = __builtin_amdgcn_wmma_f32_16x16x4_f32(
            false, a0, false, b, (short)0, acc[0][t], false, false);
        acc[1][t] = __builtin_amdgcn_wmma_f32_16x16x4_f32(
            false, a1, false, b, (short)0, acc[1][t], false, false);
      }
    }
    __syncthreads();
  }
  // store: D layout — VGPR v holds row m = v + 8*lh, col = l16
  #pragma unroll
  for (int a = 0; a < 2; ++a)
    #pragma unroll
    for (int t = 0; t < 4; ++t)
      #pragma unroll
      for (int v = 0; v < 8; ++v) {
        int m = m0 + wM + a * 16 + v + 8 * lh;
        int n = n0 + wN + t * 16 + l16;
        C[m * NDIM + n] = acc[a][t][v];
      }
}

// ---------------- overlap-add gather + odd reflection padding ----------------
__device__ __forceinline__ float sig_at(const float* __restrict__ C1, int b, int s) {
  int fmax = s >> 8; if (fmax > 63) fmax = 63;
  int fmin = (s >= WIN) ? ((s - (WIN - 1) + 255) >> 8) : 0;
  float acc = 0.f;
  for (int f = fmin; f <= fmax; ++f)
    acc += C1[(b * FRAMES + f) * NDIM + (s - (f << 8))];
  return acc;
}

__global__ __launch_bounds__(256) void build_ext(const float* __restrict__ C1,
                                                 float* __restrict__ ext_t) {
  int flat = blockIdx.x * 256 + threadIdx.x;       // 64*17186
  if (flat >= BATCH * EXT_LEN) return;
  int b = flat & 63;
  int i = flat >> 6;
  float v;
  if (i < PADL) {
    v = 2.f * sig_at(C1, b, 0) - sig_at(C1, b, PADL - i);
  } else if (i < PADL + SIG_LEN) {
    v = sig_at(C1, b, i - PADL);
  } else {
    int j = i - PADL - SIG_LEN;
    v = 2.f * sig_at(C1, b, SIG_LEN - 1) - sig_at(C1, b, SIG_LEN - 2 - j);
  }
  ext_t[i * BATCH + b] = v;   // batch-major (coalesced for IIR scan)
}

// ---------------- one DF-II-transposed step (a1=a3=a5=0) ----------------
__device__ __forceinline__ float iir_step(float x, float& z0, float& z1,
                                          float& z2, float& z3, float& z4) {
  float y = fmaf(FB0, x, z0);
  z0 = fmaf(FB1, x, z1);
  z1 = z2 + fmaf(FB2, x, -FA2 * y);
  z2 = fmaf(FB2, x, z3);
  z3 = z4 + fmaf(FB1, x, -FA4 * y);
  z4 = FB0 * x;
  return y;
}

// ---------------- forward IIR: segment-parallel (68 blocks x 64 batch-lanes) ----------------
__global__ __launch_bounds__(64) void iir_fwd(const float* __restrict__ ext_t,
                                              float* __restrict__ ybuf_t) {
  const int b  = threadIdx.x;
  const int s  = blockIdx.x;
  const int t0 = s * SEG;
  const int te = (t0 + SEG < EXT_LEN) ? (t0 + SEG) : EXT_LEN;
  float z0, z1, z2, z3, z4;
  if (s == 0) {
    float x0 = ext_t[b];
    z0 = ZI0 * x0; z1 = ZI1 * x0; z2 = ZI2 * x0; z3 = ZI3 * x0; z4 = ZI4 * x0;
  } else {
    z0 = z1 = z2 = z3 = z4 = 0.f;
    for (int t = t0 - WARM; t < t0; ++t)           // zero-state warm-up
      (void)iir_step(ext_t[t * BATCH + b], z0, z1, z2, z3, z4);
  }
  for (int t = t0; t < te; ++t)
    ybuf_t[t * BATCH + b] = iir_step(ext_t[t * BATCH + b], z0, z1, z2, z3, z4);
}

// ---------------- backward IIR over reversed y; crop+reverse into filt ----------------
__global__ __launch_bounds__(64) void iir_bwd(const float* __restrict__ ybuf_t,
                                              float* __restrict__ filt_t) {
  const int b   = threadIdx.x;
  const int s   = blockIdx.x;
  const int rt0 = s * SEG;
  const int rte = (rt0 + SEG < EXT_LEN) ? (rt0 + SEG) : EXT_LEN;
  float z0, z1, z2, z3, z4;
  if (s == 0) {
    float y0 = ybuf_t[(EXT_LEN - 1) * BATCH + b];  // zi * y[last]
    z0 = ZI0 * y0; z1 = ZI1 * y0; z2 = ZI2 * y0; z3 = ZI3 * y0; z4 = ZI4 * y0;
  } else {
    z0 = z1 = z2 = z3 = z4 = 0.f;
    for (int rt = rt0 - WARM; rt < rt0; ++rt)
      (void)iir_step(ybuf_t[(EXT_LEN - 1 - rt) * BATCH + b], z0, z1, z2, z3, z4);
  }
  for (int rt = rt0; rt < rte; ++rt) {
    float y = iir_step(ybuf_t[(EXT_LEN - 1 - rt) * BATCH + b], z0, z1, z2, z3, z4);
    int p = (EXT_LEN - 1 - rt) - PADL;             // reverse + crop
    if (p >= 0 && p < SIG_LEN) filt_t[p * BATCH + b] = y;
  }
}

// ---------------- pack frames for STFT GEMM: A2[b*64+f][n] = filt[b][f*256+n] ----------------
__global__ __launch_bounds__(256) void pack_A2(const float* __restrict__ filt_t,
                                               float* __restrict__ A2) {
  int flat = blockIdx.x * 256 + threadIdx.x;       // 4096*1024
  int r = flat >> 10, n = flat & 1023;
  int b = r >> 6, f = r & 63;
  A2[flat] = (n < WIN) ? filt_t[((f << 8) + n) * BATCH + b] : 0.f;
}

// ---------------- scatter out_mat[4096][1024] -> out [64,512,64,2] ----------------
__global__ __launch_bounds__(256) void scatter_out(const float* __restrict__ out_mat,
                                                   float* __restrict__ out) {
  int flat = blockIdx.x * 256 + threadIdx.x;       // 4,194,304
  int p = flat & 1;
  int f = (flat >> 1) & 63;
  int c = (flat >> 7) & 511;
  int b = flat >> 16;
  out[flat] = out_mat[(b * FRAMES + f) * NDIM + p * 512 + c];
}

// ---------------- launcher ----------------
extern "C" void kernel_launch(void* const* d_in, const int* in_sizes, int n_in,
                              void* d_out, int out_size, void* d_ws, size_t ws_size,
                              hipStream_t stream) {
  (void)in_sizes; (void)n_in; (void)out_size; (void)ws_size;
  const float* x = (const float*)d_in[0];
  float* out = (float*)d_out;

  float* hann  = (float*)d_ws;                       // 1024
  float* syn   = hann  + 1024;                       // 1024
  float* W1    = syn   + 1024;                       // 1M
  float* W2    = W1    + (1 << 20);                  // 1M
  float* X     = W2    + (1 << 20);                  // 4M  (also A2)
  float* C1    = X     + (4 << 20);                  // 4M  (also out_mat)
  float* ext_t = C1    + (4 << 20);                  // 17186*64
  float* ybuf  = ext_t + EXT_LEN * BATCH;            // 17186*64
  float* filt  = ybuf  + EXT_LEN * BATCH;            // 17150*64

  build_windows<<<1, 256, 0, stream>>>(hann, syn);
  build_W1<<<4096, 256, 0, stream>>>(W1, syn);
  build_W2<<<4096, 256, 0, stream>>>(W2, hann);
  pack_X<<<16384, 256, 0, stream>>>(x, X);

  dim3 gg(NDIM / 128, MROWS / 64);                   // (8, 64), 128 threads
  gemm_wmma_f32<<<gg, 128, 0, stream>>>(X, W1, C1);  // iSTFT frames (syn window folded)

  build_ext<<<(BATCH * EXT_LEN + 255) / 256, 256, 0, stream>>>(C1, ext_t);
  iir_fwd<<<NSEG, 64, 0, stream>>>(ext_t, ybuf);
  iir_bwd<<<NSEG, 64, 0, stream>>>(ybuf, filt);
  pack_A2<<<16384, 256, 0, stream>>>(filt, X);

  gemm_wmma_f32<<<gg, 128, 0, stream>>>(X, W2, C1);  // STFT (Hann folded)
  scatter_out<<<16384, 256, 0, stream>>>(C1, out);
}